// MultiLIF_17059610100001
// MI455X (gfx1250) — compile-verified
//
#include <hip/hip_runtime.h>
#include <stdint.h>

// Problem shape (fixed by reference setup_inputs()).
#define B_  32
#define L_  1024
#define K_  2048
#define KT  256          // k-neurons per block (== blockDim.x)
#define TT  64           // timesteps staged per LDS tile
#define NT  (L_ / TT)    // 16 tiles

// LIF constants
#define INV_TAU        (1.0f / 20.0f)
#define TH_            1.5f
#define INV_TAU_ADAPT  (1.0f / 100.0f)
#define BETA_ADAPT_    1.5f
#define V_RESET_       (-0.5f)

__global__ __launch_bounds__(KT) void MultiLIF_fwd_kernel(
    const float* __restrict__ I,      // (B, L, K)
    float* __restrict__ spikes,       // (B, L, K)
    float* __restrict__ vout)         // (B, L, K)
{
    // Double-buffered input staging: 2 x 64 x 256 floats = 128 KB LDS.
    __shared__ float smem[2][TT][KT];

    const int tid = threadIdx.x;                 // 0..255 -> k within tile
    const int blk = blockIdx.x;                  // 0..255
    const int b   = blk >> 3;                    // K_/KT = 8 blocks per batch row
    const int k0  = (blk & 7) * KT;

    const float* gbase = I + (size_t)b * L_ * K_ + k0;

    // Async-copy work split: each thread moves 16 B (b128) chunks.
    // Tile row (1 KB) = 64 chunks; 256 threads cover 4 rows per issue.
    const int rowsub = tid >> 6;                 // 0..3
    const int col4   = (tid & 63) * 4;           // float offset within row

    auto issue_tile = [&](int tile, int bufsel) {
        const float* gt = gbase + (size_t)(tile * TT) * K_;
        // Generic pointer to LDS: low 32 bits are the LDS byte offset.
        unsigned lbase = (unsigned)(uintptr_t)(&smem[bufsel][0][0]);
        #pragma unroll
        for (int i = 0; i < TT / 4; ++i) {       // 16 async b128 ops / thread
            const int row = i * 4 + rowsub;
            const float* gp = gt + (size_t)row * K_ + col4;
            const unsigned lp = lbase + (unsigned)((row * KT + col4) * 4);
            // ASYNCcnt-tracked DMA: global -> LDS, no VGPR data path.
            asm volatile("global_load_async_to_lds_b128 %0, %1, off"
                         :: "v"(lp), "v"(gp) : "memory");
        }
    };

    // Prologue: stage tile 0.
    issue_tile(0, 0);

    float v = 0.0f;
    float a = 0.0f;

    for (int tile = 0; tile < NT; ++tile) {
        const int cb = tile & 1;

        if (tile + 1 < NT) {
            // Refill the spare buffer (freed by the barrier at end of the
            // previous iteration), then retire this tile's 16 async ops:
            // outstanding <= 16 leaves only the newly issued tile in flight.
            issue_tile(tile + 1, cb ^ 1);
            asm volatile("s_wait_asynccnt 16" ::: "memory");
        } else {
            asm volatile("s_wait_asynccnt 0" ::: "memory");
        }
        __syncthreads();   // all waves' async portions now visible in LDS

        size_t outIdx = ((size_t)b * L_ + (size_t)tile * TT) * K_ + k0 + tid;
        #pragma unroll 8
        for (int t = 0; t < TT; ++t) {
            const float in = smem[cb][t][tid];   // ds_load_b32, conflict-free
            // Leaky integration: v += (-v/tau + i)
            v = v + (in - v * INV_TAU);
            const float th_eff = TH_ + BETA_ADAPT_ * a;
            const bool  fire   = (v >= th_eff);
            const float s      = fire ? 1.0f : 0.0f;
            // Forward value of the straight-through spike is the hard spike.
            spikes[outIdx] = s;
            vout[outIdx]   = v;                  // v_seq recorded pre-reset
            v = fire ? V_RESET_ : v;             // reset gate
            a = a + (s - a * INV_TAU_ADAPT);     // adaptation
            outIdx += K_;
        }
        __syncthreads();   // everyone done reading buf[cb] before refill
    }
}

extern "C" void kernel_launch(void* const* d_in, const int* in_sizes, int n_in,
                              void* d_out, int out_size, void* d_ws, size_t ws_size,
                              hipStream_t stream) {
    (void)in_sizes; (void)n_in; (void)out_size; (void)d_ws; (void)ws_size;

    const float* I = (const float*)d_in[0];
    float* spikes  = (float*)d_out;                       // output 0: (B,L,K)
    float* vout    = spikes + (size_t)B_ * L_ * K_;       // output 1: (B,L,K)

    dim3 grid(B_ * (K_ / KT));   // 256 blocks
    dim3 block(KT);              // 256 threads = 8 wave32
    MultiLIF_fwd_kernel<<<grid, block, 0, stream>>>(I, spikes, vout);
}